// DynamicPFNet_60009283060228
// MI455X (gfx1250) — compile-verified
//
#include <hip/hip_runtime.h>

typedef __attribute__((ext_vector_type(2))) float v2f;
typedef __attribute__((ext_vector_type(8))) float v8f;

#define PFN_VX 0.2f
#define PFN_VY 0.2f
#define PFN_X_OFF (-51.1f)   /* VX/2 - 51.2 */
#define PFN_Y_OFF (-51.1f)   /* VY/2 - 51.2 */

// Zero the x_max output region and the per-voxel accumulator (sum_x,sum_y,sum_z,count).
__global__ void pfn_zero_kernel(float* __restrict__ out, float* __restrict__ acc,
                                int n_out, int n_acc) {
    int i = blockIdx.x * blockDim.x + threadIdx.x;
    if (i < n_out) out[i] = 0.0f;
    if (i < n_acc) acc[i] = 0.0f;
}

// Per-point scatter-add of xyz and count into per-voxel accumulator.
__global__ void pfn_scatter_sum_kernel(const float* __restrict__ points,
                                       const int* __restrict__ unq_inv,
                                       float* __restrict__ acc, int N) {
    int i = blockIdx.x * blockDim.x + threadIdx.x;
    if (i >= N) return;
    int seg = unq_inv[i];
    float4 p = ((const float4*)points)[i];
    atomicAdd(&acc[seg * 4 + 0], p.x);
    atomicAdd(&acc[seg * 4 + 1], p.y);
    atomicAdd(&acc[seg * 4 + 2], p.z);
    atomicAdd(&acc[seg * 4 + 3], 1.0f);
}

// Per-voxel finalize: sums -> means (one IEEE divide per voxel, like the
// reference's segment_sum / cnt). Keeps the divide out of the 2M-point loop.
__global__ void pfn_mean_kernel(float* __restrict__ acc, int V) {
    int i = blockIdx.x * blockDim.x + threadIdx.x;
    if (i >= V) return;
    float4 s = ((const float4*)acc)[i];
    float4 m;
    m.x = s.x / s.w;
    m.y = s.y / s.w;
    m.z = s.z / s.w;
    m.w = s.w;
    ((float4*)acc)[i] = m;
}

// Main kernel: one wave32 per 16-point tile. Features (9, padded to K=12) times
// W (9x64, padded) via 12x V_WMMA_F32_16X16X4_F32, then ReLU + atomicMax scatter.
__global__ void pfn_wmma_kernel(const float* __restrict__ points,
                                const int*   __restrict__ grid_ind,
                                const int*   __restrict__ unq_inv,
                                const float* __restrict__ W,     // 9 x 64 row-major
                                const float* __restrict__ acc,   // V x 4 (means)
                                float*       __restrict__ out,   // V x 64
                                int N, int numTiles) {
    const int lane  = threadIdx.x & 31;
    const int wave  = threadIdx.x >> 5;
    const int wgWaves = blockDim.x >> 5;
    const int waveGlobal = blockIdx.x * wgWaves + wave;
    const int waveStride = gridDim.x * wgWaves;

    const int  m  = lane & 15;       // matrix row (point within tile)
    const bool hi = (lane >= 16);    // lane half selects K sub-pair
    const int  n  = lane & 15;       // output column within 16-wide tile

    // Loop-invariant bpermute byte indices: src lane = r + (hi ? 8 : 0).
    int bidx[8];
    #pragma unroll
    for (int r = 0; r < 8; ++r) bidx[r] = (r + (hi ? 8 : 0)) << 2;

    // Preload B fragments (registers, reused across all tiles).
    // B layout (4x16 f32): VGPR0 lanes0-15 -> K=ks+0, lanes16-31 -> K=ks+2;
    //                      VGPR1 lanes0-15 -> K=ks+1, lanes16-31 -> K=ks+3.
    v2f Bf[4][3];
    {
        const int khalf = hi ? 2 : 0;
        for (int t = 0; t < 4; ++t) {
            const int col = t * 16 + n;
            for (int s = 0; s < 3; ++s) {
                const int kr = s * 4 + khalf;
                v2f b;
                b.x = (kr     < 9) ? W[kr * 64 + col]       : 0.0f;
                b.y = (kr + 1 < 9) ? W[(kr + 1) * 64 + col] : 0.0f;
                Bf[t][s] = b;
            }
        }
    }

    for (int tile = waveGlobal; tile < numTiles; tile += waveStride) {
        const int  p     = tile * 16 + m;
        const bool valid = (p < N);
        const int  pi    = valid ? p : 0;

        float4 pt  = ((const float4*)points)[pi];
        int    seg = unq_inv[pi];
        int4   g   = ((const int4*)grid_ind)[pi];   // g.z = gy, g.w = gx
        float4 mu  = ((const float4*)acc)[seg];     // per-voxel xyz mean

        // 9 features (rows with invalid points produce garbage-but-finite
        // values; they only affect their own D row, which is never scattered).
        const float f0 = pt.x, f1 = pt.y, f2 = pt.z, f3 = pt.w;
        const float f4 = pt.x - mu.x;
        const float f5 = pt.y - mu.y;
        const float f6 = pt.z - mu.z;
        const float f7 = pt.x - ((float)g.w * PFN_VX + PFN_X_OFF);
        const float f8 = pt.y - ((float)g.z * PFN_VY + PFN_Y_OFF);

        // A layout (16x4 f32): lanes 0-15 row M=lane hold K={ks, ks+1};
        //                      lanes 16-31 row M=lane-16 hold K={ks+2, ks+3}.
        v2f A0, A1, A2;
        A0.x = hi ? f2 : f0;   A0.y = hi ? f3 : f1;
        A1.x = hi ? f6 : f4;   A1.y = hi ? f7 : f5;
        A2.x = hi ? 0.0f : f8; A2.y = 0.0f;        // K=8 real, K=9..11 pad

        // Segment index for each output row M = r + (hi ? 8 : 0).
        const int myseg = valid ? seg : -1;
        int rseg[8];
        #pragma unroll
        for (int r = 0; r < 8; ++r)
            rseg[r] = __builtin_amdgcn_ds_bpermute(bidx[r], myseg);

        v8f C[4];
        #pragma unroll
        for (int t = 0; t < 4; ++t) C[t] = {};

        // D = A x B + C; EXEC is all-ones at every WMMA (no divergence above).
        #pragma unroll
        for (int t = 0; t < 4; ++t) {
            C[t] = __builtin_amdgcn_wmma_f32_16x16x4_f32(
                false, A0, false, Bf[t][0], (short)0, C[t], false, false);
            C[t] = __builtin_amdgcn_wmma_f32_16x16x4_f32(
                false, A1, false, Bf[t][1], (short)0, C[t], false, false);
            C[t] = __builtin_amdgcn_wmma_f32_16x16x4_f32(
                false, A2, false, Bf[t][2], (short)0, C[t], false, false);
        }

        // Segment max with implicit ReLU: out is 0-initialized and relu>=0,
        // so only v>0 contributions matter; int-bit atomicMax is order-correct
        // for non-negative floats. One base address per row; the t*16 column
        // step (64 bytes) folds into the atomic's immediate offset.
        #pragma unroll
        for (int r = 0; r < 8; ++r) {
            const int sgt = rseg[r];
            if (sgt >= 0) {
                int* base = (int*)out + ((size_t)sgt * 64 + n);
                #pragma unroll
                for (int t = 0; t < 4; ++t) {
                    const float v = C[t][r];
                    if (v > 0.0f) {
                        atomicMax(base + t * 16, __float_as_int(v));
                    }
                }
            }
        }
    }
}

extern "C" void kernel_launch(void* const* d_in, const int* in_sizes, int n_in,
                              void* d_out, int out_size, void* d_ws, size_t ws_size,
                              hipStream_t stream) {
    const float* points   = (const float*)d_in[0];
    const int*   grid_ind = (const int*)d_in[1];
    const int*   unq      = (const int*)d_in[2];
    const int*   unq_inv  = (const int*)d_in[3];
    const float* W        = (const float*)d_in[4];

    const int N = in_sizes[0] / 4;
    const int V = in_sizes[2] / 4;

    float* out = (float*)d_out;   // V*64 x_max, then V*4 raw int32 (unq)
    float* acc = (float*)d_ws;    // V*4: sums then (after finalize) means

    // 1) zero output + accumulator (every call: atomics accumulate otherwise)
    {
        int nzero = V * 64;
        int blocks = (nzero + 255) / 256;
        pfn_zero_kernel<<<blocks, 256, 0, stream>>>(out, acc, nzero, V * 4);
    }

    // 2) per-voxel xyz sums + counts
    {
        int blocks = (N + 255) / 256;
        pfn_scatter_sum_kernel<<<blocks, 256, 0, stream>>>(points, unq_inv, acc, N);
    }

    // 3) per-voxel mean finalize (one divide per voxel)
    {
        int blocks = (V + 255) / 256;
        pfn_mean_kernel<<<blocks, 256, 0, stream>>>(acc, V);
    }

    // 4) features -> WMMA GEMM -> relu -> segment max
    {
        int numTiles = (N + 15) / 16;
        int wavesPerBlock = 8;                       // 256 threads = 8 wave32
        int want = (numTiles + wavesPerBlock - 1) / wavesPerBlock;
        int blocks = want < 4096 ? want : 4096;      // grid-stride persistent waves
        if (blocks < 1) blocks = 1;
        pfn_wmma_kernel<<<blocks, 256, 0, stream>>>(points, grid_ind, unq_inv, W,
                                                    acc, out, N, numTiles);
    }

    // 5) pass-through unq (int32 bits) into the tail of d_out
    hipMemcpyAsync((char*)d_out + (size_t)V * 64 * sizeof(float), unq,
                   (size_t)V * 4 * sizeof(int), hipMemcpyDeviceToDevice, stream);
}